// TDAG_Structure2Vec_13958643712644
// MI455X (gfx1250) — compile-verified
//
#include <hip/hip_runtime.h>

typedef __attribute__((ext_vector_type(2))) float v2f;
typedef __attribute__((ext_vector_type(8))) float v8f;

#define CH 128
#define LDSS 130   // padded LDS row stride (floats) to avoid bank conflicts

// mu = relu( X@W1^T  [+ MIN@W2^T + MOUT@W3^T] )
// Block: 256 threads = 8 waves. Block handles 16 rows (M-tile), wave w handles
// columns [16w, 16w+16) (N-tile). fp32 WMMA 16x16x4, K-loop over 128 in steps of 4.
__global__ void __launch_bounds__(256)
s2v_update(const float* __restrict__ X,
           const float* __restrict__ MIN,
           const float* __restrict__ MOUT,
           const float* __restrict__ W1,
           const float* __restrict__ W2,
           const float* __restrict__ W3,
           float* __restrict__ MU,
           int use_msgs)
{
    __shared__ float sX[16 * LDSS];
    __shared__ float sI[16 * LDSS];
    __shared__ float sO[16 * LDSS];

    const int row0 = blockIdx.x << 4;
    const int tid  = threadIdx.x;

    // Cooperative, coalesced stage of the three 16x128 A-tiles into LDS.
    for (int i = tid; i < 16 * CH; i += 256) {
        const int r = i >> 7;
        const int c = i & (CH - 1);
        const size_t g = (size_t)(row0 + r) * CH + c;
        sX[r * LDSS + c] = X[g];
        if (use_msgs) {
            sI[r * LDSS + c] = MIN[g];
            sO[r * LDSS + c] = MOUT[g];
        }
    }
    __syncthreads();

    const int lane  = tid & 31;
    const int col0  = (tid >> 5) << 4;     // wave's N-tile base
    const int nloc  = lane & 15;           // M index for A-frag, N index for B/D
    const int khalf = (lane >> 4) << 1;    // 0 (lanes 0-15) or 2 (lanes 16-31)

    v8f acc = {};

    // GEMM 1: A = x rows, B^T = W1 rows (W is [out,in] row-major -> B[k][n] = W[n][k])
    {
        const float* wr = W1 + (size_t)(col0 + nloc) * CH + khalf;
        const float* ar = sX + nloc * LDSS + khalf;
        #pragma unroll
        for (int kk = 0; kk < 32; ++kk) {
            const int k0 = kk << 2;
            v2f a, b;
            a.x = ar[k0]; a.y = ar[k0 + 1];
            b.x = wr[k0]; b.y = wr[k0 + 1];
            acc = __builtin_amdgcn_wmma_f32_16x16x4_f32(
                false, a, false, b, (short)0, acc, false, false);
        }
    }
    if (use_msgs) {
        const float* wr = W2 + (size_t)(col0 + nloc) * CH + khalf;
        const float* ar = sI + nloc * LDSS + khalf;
        #pragma unroll
        for (int kk = 0; kk < 32; ++kk) {
            const int k0 = kk << 2;
            v2f a, b;
            a.x = ar[k0]; a.y = ar[k0 + 1];
            b.x = wr[k0]; b.y = wr[k0 + 1];
            acc = __builtin_amdgcn_wmma_f32_16x16x4_f32(
                false, a, false, b, (short)0, acc, false, false);
        }
        const float* wr3 = W3 + (size_t)(col0 + nloc) * CH + khalf;
        const float* ar3 = sO + nloc * LDSS + khalf;
        #pragma unroll
        for (int kk = 0; kk < 32; ++kk) {
            const int k0 = kk << 2;
            v2f a, b;
            a.x = ar3[k0]; a.y = ar3[k0 + 1];
            b.x = wr3[k0]; b.y = wr3[k0 + 1];
            acc = __builtin_amdgcn_wmma_f32_16x16x4_f32(
                false, a, false, b, (short)0, acc, false, false);
        }
    }

    // D layout: VGPR v -> lanes 0-15: (M=v, N=lane); lanes 16-31: (M=v+8, N=lane-16)
    const int mbase = (lane >> 4) << 3;
    #pragma unroll
    for (int v = 0; v < 8; ++v) {
        float val = acc[v];
        val = val > 0.0f ? val : 0.0f;   // relu
        MU[(size_t)(row0 + mbase + v) * CH + col0 + nloc] = val;
    }
}

// Zero a contiguous float buffer (count given in float4 units).
__global__ void s2v_zero(float4* __restrict__ p, int n4)
{
    const int stride = gridDim.x * blockDim.x;
    const float4 z = {0.f, 0.f, 0.f, 0.f};
    for (int i = blockIdx.x * blockDim.x + threadIdx.x; i < n4; i += stride)
        p[i] = z;
}

// One wave (32 lanes) per edge; lane handles 4 consecutive channels (float4).
// msg_in[dst] += mu[src]; msg_out[src] += mu[dst]
__global__ void s2v_scatter(const int* __restrict__ src, const int* __restrict__ dst,
                            const float* __restrict__ mu,
                            float* __restrict__ msg_in, float* __restrict__ msg_out,
                            int n_edges)
{
    const int gid = blockIdx.x * blockDim.x + threadIdx.x;
    const int e = gid >> 5;
    if (e >= n_edges) return;
    const int lane = gid & 31;
    const int s = src[e];
    const int d = dst[e];
    const int c = lane << 2;

    const float4 vs = *(const float4*)(mu + (size_t)s * CH + c);
    const float4 vd = *(const float4*)(mu + (size_t)d * CH + c);

    float* pin = msg_in  + (size_t)d * CH + c;
    float* pot = msg_out + (size_t)s * CH + c;
    atomicAdd(pin + 0, vs.x); atomicAdd(pin + 1, vs.y);
    atomicAdd(pin + 2, vs.z); atomicAdd(pin + 3, vs.w);
    atomicAdd(pot + 0, vd.x); atomicAdd(pot + 1, vd.y);
    atomicAdd(pot + 2, vd.z); atomicAdd(pot + 3, vd.w);
}

__global__ void s2v_out_zero(float* __restrict__ out)
{
    out[threadIdx.x] = 0.0f;
}

// h_G = sum over nodes of mu. Block = 128 threads (one per channel), 256 rows/block.
__global__ void s2v_reduce(const float* __restrict__ mu, float* __restrict__ out,
                           int n_nodes)
{
    const int t  = threadIdx.x;
    const int r0 = blockIdx.x * 256;
    int rend = r0 + 256;
    if (rend > n_nodes) rend = n_nodes;
    float acc = 0.0f;
    for (int r = r0; r < rend; ++r)
        acc += mu[(size_t)r * CH + t];
    atomicAdd(&out[t], acc);
}

extern "C" void kernel_launch(void* const* d_in, const int* in_sizes, int n_in,
                              void* d_out, int out_size, void* d_ws, size_t ws_size,
                              hipStream_t stream)
{
    const float* X  = (const float*)d_in[0];   // x_log [N,128]
    const int*   EI = (const int*)  d_in[1];   // edge_index [2, E]
    const float* W1 = (const float*)d_in[2];   // [128,128] row-major [out,in]
    const float* W2 = (const float*)d_in[3];
    const float* W3 = (const float*)d_in[4];

    const int n_nodes = in_sizes[0] / CH;
    const int n_edges = in_sizes[1] / 2;
    const int* src = EI;
    const int* dst = EI + n_edges;

    float* mu      = (float*)d_ws;
    float* msg_in  = mu     + (size_t)n_nodes * CH;
    float* msg_out = msg_in + (size_t)n_nodes * CH;
    float* out     = (float*)d_out;

    const int ublocks = n_nodes / 16;                     // 6250 (N is a multiple of 16)
    const int sblocks = (n_edges * 32 + 255) / 256;       // 32 lanes per edge
    const int msg4    = (2 * n_nodes * CH) / 4;           // msg_in & msg_out contiguous

    // Iteration 1: mu = relu(X @ W1^T)  (messages are zero -> skip scatter)
    s2v_update<<<ublocks, 256, 0, stream>>>(X, X, X, W1, W2, W3, mu, 0);

    // Iterations 2 and 3
    for (int it = 0; it < 2; ++it) {
        s2v_zero<<<2048, 256, 0, stream>>>((float4*)msg_in, msg4);
        s2v_scatter<<<sblocks, 256, 0, stream>>>(src, dst, mu, msg_in, msg_out, n_edges);
        s2v_update<<<ublocks, 256, 0, stream>>>(X, msg_in, msg_out, W1, W2, W3, mu, 1);
    }

    // h_G = column sum of mu
    s2v_out_zero<<<1, CH, 0, stream>>>(out);
    s2v_reduce<<<(n_nodes + 255) / 256, CH, 0, stream>>>(mu, out, n_nodes);
}